// CSRSparseRetrievalModelIterative_57947698758426
// MI455X (gfx1250) — compile-verified
//
#include <hip/hip_runtime.h>
#include <stdint.h>

// ---------------------------------------------------------------------------
// Sparse CSC retrieval: scatter-add 64 posting lists (1600 each) into a
// 1M-float accumulator (L2-resident: 4MB << 192MB), then top-10 of 1M.
// Memory/launch-bound (~9MB traffic total @ 23.3TB/s HBM, mostly L2 hits).
// CDNA5 paths used: global_load_async_to_lds_b128 + s_wait_asynccnt
// (double-buffered streaming scan), ds_* merge trees, global_atomic_add_f32.
// WMMA is not applicable: no dense matmul exists in this workload.
// ---------------------------------------------------------------------------

#define TOPK        10
#define TPB         256
#define NB_TOPK     256          // blocks in the block-topk scan
#define TILE_FLOATS 1024         // 256 threads * 4 floats (b128 per lane) = 4KB

__device__ __forceinline__ bool better(float a, int ia, float b, int ib) {
    // descending by value; ties -> lower index (matches jax.lax.top_k)
    return (a > b) || (a == b && ia < ib);
}

// Fully-unrolled sorted-insert into a register-resident descending top-K list.
__device__ __forceinline__ void insert_top(float v, int id,
                                           float (&val)[TOPK], int (&idx)[TOPK]) {
    if (!better(v, id, val[TOPK - 1], idx[TOPK - 1])) return;
    val[TOPK - 1] = v; idx[TOPK - 1] = id;
#pragma unroll
    for (int k = TOPK - 1; k > 0; --k) {
        if (better(val[k], idx[k], val[k - 1], idx[k - 1])) {
            float tv = val[k];  val[k] = val[k - 1];  val[k - 1] = tv;
            int   ti = idx[k];  idx[k] = idx[k - 1];  idx[k - 1] = ti;
        }
    }
}

// LDS tree-merge of TPB sorted TOPK-lists down to one list at slot 0.
__device__ __forceinline__ void tree_merge(float* s_val, int* s_idx, int tid) {
    for (int half = TPB / 2; half >= 1; half >>= 1) {
        __syncthreads();
        if (tid < half) {
            const int a = tid * TOPK;
            const int b = (tid + half) * TOPK;
            int i = 0, j = 0;
            float tv[TOPK]; int ti[TOPK];
#pragma unroll
            for (int k = 0; k < TOPK; ++k) {      // i+j==k, so i,j<=9 on reads
                float av = s_val[a + i]; int ai = s_idx[a + i];
                float bv = s_val[b + j]; int bi = s_idx[b + j];
                if (better(av, ai, bv, bi)) { tv[k] = av; ti[k] = ai; ++i; }
                else                        { tv[k] = bv; ti[k] = bi; ++j; }
            }
#pragma unroll
            for (int k = 0; k < TOPK; ++k) { s_val[a + k] = tv[k]; s_idx[a + k] = ti[k]; }
        }
    }
    __syncthreads();
}

// ---------------------------------------------------------------------------
// Vectorized zero of acc[n_docs]: b128 stores for the bulk, scalar tail.
__global__ void k_zero(float* __restrict__ acc, const int* __restrict__ n_docs_p) {
    const int n  = *n_docs_p;
    const int n4 = n >> 2;
    float4 z4 = make_float4(0.f, 0.f, 0.f, 0.f);
    float4* __restrict__ a4 = (float4*)acc;
    for (int i = blockIdx.x * blockDim.x + threadIdx.x; i < n4;
         i += gridDim.x * blockDim.x)
        a4[i] = z4;                                  // global_store_b128
    // scalar tail (n not multiple of 4)
    int t = (n4 << 2) + blockIdx.x * blockDim.x + threadIdx.x;
    if (t < n) acc[t] = 0.0f;
}

// 2D grid: y = query term (no div/mod), x strides the postings of that column.
// rind/cval loads are perfectly coalesced; qidx/qval/ccol become scalar loads.
__global__ void k_scatter(const int* __restrict__ qidx, const float* __restrict__ qval,
                          const int* __restrict__ ccol, const int* __restrict__ rind,
                          const float* __restrict__ cval,
                          float* __restrict__ acc) {
    const int   q   = blockIdx.y;
    const int   col = qidx[q];                       // scalar (uniform in block)
    const int   st  = ccol[col];
    const int   len = ccol[col + 1] - st;
    const float qv  = qval[q];
    for (int j = blockIdx.x * blockDim.x + threadIdx.x; j < len;
         j += gridDim.x * blockDim.x) {
        const int pos = st + j;
        const int doc = __builtin_nontemporal_load(rind + pos);   // streamed once
        const float v = __builtin_nontemporal_load(cval + pos) * qv;
        atomicAdd(acc + doc, v);                     // global_atomic_add_f32 -> L2
    }
}

// Per-block top-K over a contiguous chunk of acc.  Full tiles are streamed
// through LDS with double-buffered CDNA5 async copies (ASYNCcnt path).
__global__ void k_block_topk(const float* __restrict__ acc,
                             const int* __restrict__ n_docs_p,
                             float* __restrict__ cand_val, int* __restrict__ cand_idx) {
    __shared__ float s_tile[2][TILE_FLOATS];
    __shared__ float s_val[TPB * TOPK];
    __shared__ int   s_idx[TPB * TOPK];

    const int tid   = threadIdx.x;
    const int n     = *n_docs_p;
    const int chunk = (n + (int)gridDim.x - 1) / (int)gridDim.x;
    const int cbeg  = (int)blockIdx.x * chunk;
    const int cend  = (cbeg + chunk < n) ? (cbeg + chunk) : n;

    float val[TOPK]; int idx[TOPK];
#pragma unroll
    for (int k = 0; k < TOPK; ++k) { val[k] = -3.402823466e38f; idx[k] = 0x7fffffff; }

    const int nfull = (cend > cbeg) ? (cend - cbeg) / TILE_FLOATS : 0;

    auto issue = [&](int buf, int tile) {
        // Per-lane 16B: LDS dest offset in a 32-bit VGPR, 64-bit global vaddr, GV mode.
        unsigned     lds_off = (unsigned)(uintptr_t)&s_tile[buf][tid * 4];
        const float* g       = acc + cbeg + tile * TILE_FLOATS + tid * 4;
        asm volatile("global_load_async_to_lds_b128 %0, %1, off"
                     :: "v"(lds_off), "v"(g) : "memory");
    };

    if (nfull > 0) issue(0, 0);
    for (int t = 0; t < nfull; ++t) {
        if (t + 1 < nfull) {
            issue((t + 1) & 1, t + 1);                       // prefetch next tile
            asm volatile("s_wait_asynccnt 1" ::: "memory");  // tile t ready
        } else {
            asm volatile("s_wait_asynccnt 0" ::: "memory");
        }
        __syncthreads();
        const float* tp   = s_tile[t & 1];
        const int    base = cbeg + t * TILE_FLOATS;
#pragma unroll
        for (int e = 0; e < TILE_FLOATS / TPB; ++e) {
            const int li = e * TPB + tid;
            insert_top(tp[li], base + li, val, idx);         // ds_load_b32 reads
        }
        __syncthreads();   // everyone done reading before buffer reuse
    }
    // tail (partial tile): direct NT global loads with bounds check
    for (int i = cbeg + nfull * TILE_FLOATS + tid; i < cend; i += TPB)
        insert_top(__builtin_nontemporal_load(acc + i), i, val, idx);

#pragma unroll
    for (int k = 0; k < TOPK; ++k) { s_val[tid * TOPK + k] = val[k]; s_idx[tid * TOPK + k] = idx[k]; }
    tree_merge(s_val, s_idx, tid);

    if (tid < TOPK) {
        cand_val[blockIdx.x * TOPK + tid] = s_val[tid];
        cand_idx[blockIdx.x * TOPK + tid] = s_idx[tid];
    }
}

__global__ void k_final_topk(const float* __restrict__ cand_val,
                             const int* __restrict__ cand_idx,
                             int ncand, float* __restrict__ out) {
    __shared__ float s_val[TPB * TOPK];
    __shared__ int   s_idx[TPB * TOPK];
    const int tid = threadIdx.x;

    float val[TOPK]; int idx[TOPK];
#pragma unroll
    for (int k = 0; k < TOPK; ++k) { val[k] = -3.402823466e38f; idx[k] = 0x7fffffff; }

    for (int i = tid; i < ncand; i += TPB)
        insert_top(cand_val[i], cand_idx[i], val, idx);

#pragma unroll
    for (int k = 0; k < TOPK; ++k) { s_val[tid * TOPK + k] = val[k]; s_idx[tid * TOPK + k] = idx[k]; }
    tree_merge(s_val, s_idx, tid);

    if (tid == 0) {
#pragma unroll
        for (int k = 0; k < TOPK; ++k) {
            out[k]        = s_val[k];            // top_vals
            out[TOPK + k] = (float)s_idx[k];     // top_idx (float-typed d_out)
        }
    }
}

// ---------------------------------------------------------------------------
extern "C" void kernel_launch(void* const* d_in, const int* in_sizes, int n_in,
                              void* d_out, int out_size, void* d_ws, size_t ws_size,
                              hipStream_t stream) {
    const int*   qidx     = (const int*)  d_in[0];   // [1,Q] int32
    const float* qval     = (const float*)d_in[1];   // [1,Q] f32
    const int*   ccol     = (const int*)  d_in[2];   // [VOCAB+1]
    const int*   rind     = (const int*)  d_in[3];   // [nnz]
    const float* cval     = (const float*)d_in[4];   // [nnz]
    const int*   n_docs_p = (const int*)  d_in[5];   // scalar on device
    // d_in[6] = nnz_max (device scalar; host estimate derived from sizes)
    // d_in[7] = top_k (compile-time TOPK = 10)

    const int Q     = in_sizes[1];
    const int vocab = in_sizes[2] - 1;
    const int nnz   = in_sizes[4];
    const int nnz_max_host = (vocab > 0) ? nnz / vocab : 1;  // grid sizing only

    // Workspace layout: acc[n_docs] at offset 0; candidate lists carved from
    // the END of the workspace (host never needs n_docs).
    float* acc = (float*)d_ws;
    const size_t cand_bytes = (size_t)NB_TOPK * TOPK * (sizeof(float) + sizeof(int));
    size_t cand_off = (ws_size > cand_bytes + 256) ? ((ws_size - cand_bytes) & ~(size_t)255) : 0;
    float* cand_val = (float*)((char*)d_ws + cand_off);
    int*   cand_idx = (int*)(cand_val + NB_TOPK * TOPK);

    k_zero<<<1024, TPB, 0, stream>>>(acc, n_docs_p);

    // 2D scatter grid: x covers postings of one column, y = query term.
    int nbx = (nnz_max_host + TPB - 1) / TPB;        // 1600/256 -> 7
    if (nbx < 1) nbx = 1;
    if (nbx > 1024) nbx = 1024;                      // grid-stride covers rest
    dim3 sc_grid(nbx, (unsigned)(Q > 0 ? Q : 1), 1);
    k_scatter<<<sc_grid, TPB, 0, stream>>>(qidx, qval, ccol, rind, cval, acc);

    k_block_topk<<<NB_TOPK, TPB, 0, stream>>>(acc, n_docs_p, cand_val, cand_idx);
    k_final_topk<<<1, TPB, 0, stream>>>(cand_val, cand_idx, NB_TOPK * TOPK, (float*)d_out);
}